// MultiHeadAttention_76166950028300
// MI455X (gfx1250) — compile-verified
//
#include <hip/hip_runtime.h>
#include <hip/hip_bf16.h>

// MultiHeadAttention forward for MI455X (gfx1250), fp32 via V_WMMA_F32_16X16X4_F32.
// d_ws layout: Q[B*H*T*HD] | K[..] | V[..] | ctx[T*B*E]  (4 x 16 MB = 64 MB)

typedef __attribute__((ext_vector_type(2))) float v2f;
typedef __attribute__((ext_vector_type(8))) float v8f;

#define T_   2048
#define B_   2
#define E_   1024
#define H_   16
#define HD_  64
#define SCALE_ 0.125f      // HD^-0.5
#define SWP  2050          // padded LDS row stride (2050 % 64 == 2, even -> b64-aligned, conflict-free)

#define KC     32          // GEMM K-chunk
#define ASTR   34          // As row stride (pad 2): 34*l mod 64 distinct, even
#define BSTR   132         // Bs k-pair stride: 2*64 + 4 pad, even

static __device__ __forceinline__ v8f wmma_f32(v2f a, v2f b, v8f c) {
  // D = A(16x4 f32) * B(4x16 f32) + C(16x16 f32)
  return __builtin_amdgcn_wmma_f32_16x16x4_f32(false, a, false, b, (short)0, c, false, false);
}

// C[m,n] = sum_k A[m,k]*W[n,k] + bias[n]
// mode 0: write Co[m*N+n]   (out projection)
// mode 1: scatter into per-head Q(*SCALE)/K/V buffers  (QKV projection)
__global__ __launch_bounds__(128) void gemm_xwT_kernel(
    const float* __restrict__ A, const float* __restrict__ W,
    const float* __restrict__ bias,
    float* __restrict__ Co,
    float* __restrict__ Qd, float* __restrict__ Kd, float* __restrict__ Vd,
    int N, int K, int mode)
{
  __shared__ __align__(16) float As[64 * ASTR];        // A tile [64][KC], padded rows
  __shared__ __align__(16) float Bs[(KC / 2) * BSTR];  // B tile k-pair interleaved: Bs[kp][n][2]

  const int tid  = threadIdx.x;
  const int lane = tid & 31, wave = tid >> 5;
  const int half = lane >> 4, lr = lane & 15;
  const int m0 = blockIdx.y * 64;
  const int n0 = blockIdx.x * 64;
  const int rowOff = (wave >> 1) * 32;
  const int colOff = (wave & 1) * 32;

  v8f c00 = {}, c01 = {}, c10 = {}, c11 = {};

  for (int k0 = 0; k0 < K; k0 += KC) {
    for (int idx = tid; idx < 64 * KC; idx += 128) {   // A: coalesced 128B rows
      int r = idx >> 5, kk = idx & (KC - 1);
      As[r * ASTR + kk] = A[(size_t)(m0 + r) * K + k0 + kk];
    }
    for (int idx = tid; idx < 64 * KC; idx += 128) {   // W: coalesced read, pair-interleaved store
      int n = idx >> 5, k = idx & (KC - 1);
      Bs[(k >> 1) * BSTR + n * 2 + (k & 1)] = W[(size_t)(n0 + n) * K + k0 + k];
    }
    __syncthreads();
#pragma unroll
    for (int kk = 0; kk < KC; kk += 4) {
      const int ao = kk + 2 * half;                    // A K-pair for this lane half
      const int kp = (kk >> 1) + half;                 // B k-pair index
      v2f a0 = *(const v2f*)&As[(rowOff + lr) * ASTR + ao];
      v2f a1 = *(const v2f*)&As[(rowOff + 16 + lr) * ASTR + ao];
      v2f b0 = *(const v2f*)&Bs[kp * BSTR + (colOff + lr) * 2];        // single b64, no movs
      v2f b1 = *(const v2f*)&Bs[kp * BSTR + (colOff + 16 + lr) * 2];
      c00 = wmma_f32(a0, b0, c00);
      c01 = wmma_f32(a0, b1, c01);
      c10 = wmma_f32(a1, b0, c10);
      c11 = wmma_f32(a1, b1, c11);
    }
    __syncthreads();
  }

#pragma unroll
  for (int sub = 0; sub < 4; ++sub) {
    const v8f& cc = (sub == 0) ? c00 : (sub == 1) ? c01 : (sub == 2) ? c10 : c11;
    const int ri = (sub >> 1) * 16;
    const int cj = (sub & 1) * 16;
    const int gn = n0 + colOff + cj + lr;
    const float bv = bias[gn];
#pragma unroll
    for (int i = 0; i < 8; ++i) {
      const int gm = m0 + rowOff + ri + i + 8 * half;  // C/D: VGPR i -> row i (+8 for hi lanes)
      float val = cc[i] + bv;
      if (mode == 1) {
        const int which = gn >> 10;        // /E_: 0=q 1=k 2=v
        const int e = gn & 1023;
        if (which == 0) val *= SCALE_;
        const int hh = e >> 6, d = e & 63;
        const int t = gm >> 1, bb = gm & 1;            // row = t*B + b, B=2
        float* dst = (which == 0) ? Qd : (which == 1) ? Kd : Vd;
        dst[(((size_t)(bb * H_ + hh) * T_) + t) * HD_ + d] = val;
      } else {
        Co[(size_t)gm * N + gn] = val;
      }
    }
  }
}

// One block per (batch, 16-row strip); loops all 16 heads.
// Scores+softmax in LDS, K/V streamed from global (L2-resident: whole QKV ~50MB << 192MB L2).
__global__ __launch_bounds__(256) void attn_kernel(
    const float* __restrict__ Qh, const float* __restrict__ Kh,
    const float* __restrict__ Vh,
    float* __restrict__ ctx, float* __restrict__ avgW)
{
  extern __shared__ float sm[];
  float* Sw  = sm;               // [16][SWP] score/weight strip
  float* acc = sm + 16 * SWP;    // [16][64]  attn accumulator

  const int tid  = threadIdx.x;
  const int lane = tid & 31, wave = tid >> 5;
  const int half = lane >> 4, lr = lane & 15;
  const int bb = blockIdx.x >> 7;            // batch
  const int t0 = (blockIdx.x & 127) * 16;    // row strip base
  const int r   = tid >> 4;                  // softmax row (0..15), 16 threads/row
  const int c16 = tid & 15;

  for (int h = 0; h < H_; ++h) {
    const float* Q  = Qh + (((size_t)(bb * H_ + h) * T_) + t0) * HD_;
    const float* Kp = Kh + (size_t)(bb * H_ + h) * T_ * HD_;
    const float* Vp = Vh + (size_t)(bb * H_ + h) * T_ * HD_;

    for (int idx = tid; idx < 16 * 64; idx += 256) acc[idx] = 0.f;   // barrier below covers this

    // Q A-fragments for the strip: lane half 0 -> K pair (4k,4k+1), half 1 -> (4k+2,4k+3)
    v2f qa[16];
#pragma unroll
    for (int kk = 0; kk < 16; ++kk)
      qa[kk] = *(const v2f*)(Q + lr * HD_ + kk * 4 + 2 * half);

    // ---- scores S[16 x 2048] = Q Khat^T ----
    for (int st = wave; st < T_ / 16; st += 8) {
      const int s0 = st * 16;
      v8f c = {};
#pragma unroll
      for (int kk = 0; kk < 16; ++kk) {
        // B[k][n] = K[s0+n][k]: row-major K gives contiguous b64 per lane
        v2f kb = *(const v2f*)(Kp + (size_t)(s0 + lr) * HD_ + kk * 4 + 2 * half);
        c = wmma_f32(qa[kk], kb, c);
      }
#pragma unroll
      for (int i = 0; i < 8; ++i)
        Sw[(i + 8 * half) * SWP + s0 + lr] = c[i];
    }
    __syncthreads();

    // ---- softmax (row r owned by 16 consecutive lanes -> shfl_xor reduce) ----
    float mx = -3.4e38f;
    for (int j = c16; j < T_; j += 16) mx = fmaxf(mx, Sw[r * SWP + j]);
#pragma unroll
    for (int o = 8; o >= 1; o >>= 1) mx = fmaxf(mx, __shfl_xor(mx, o, 32));
    float sum = 0.f;
    for (int j = c16; j < T_; j += 16) {
      float e = __expf(Sw[r * SWP + j] - mx);
      Sw[r * SWP + j] = e;
      sum += e;
    }
#pragma unroll
    for (int o = 8; o >= 1; o >>= 1) sum += __shfl_xor(sum, o, 32);
    const float inv = 1.f / sum;
    float* gw = avgW + (size_t)bb * T_ * T_ + (size_t)(t0 + r) * T_;
    for (int j = c16; j < T_; j += 16) {
      float w = Sw[r * SWP + j] * inv;
      Sw[r * SWP + j] = w;
      float contrib = w * (1.f / H_);        // head-averaged weights, same thread each head
      if (h == 0) gw[j] = contrib; else gw[j] += contrib;
    }
    __syncthreads();

    // ---- attn strip = W[16 x 2048] @ V[2048 x 64], s-range split across 8 waves ----
    v8f pc0 = {}, pc1 = {}, pc2 = {}, pc3 = {};
    const int sBase = wave * (T_ / 8);
    for (int k4 = 0; k4 < (T_ / 8) / 4; ++k4) {
      const int kk = sBase + k4 * 4 + 2 * half;
      v2f wa = *(const v2f*)&Sw[lr * SWP + kk];
      v2f vb;
      vb.x = Vp[(size_t)(kk + 0) * HD_ +  0 + lr]; vb.y = Vp[(size_t)(kk + 1) * HD_ +  0 + lr];
      pc0 = wmma_f32(wa, vb, pc0);
      vb.x = Vp[(size_t)(kk + 0) * HD_ + 16 + lr]; vb.y = Vp[(size_t)(kk + 1) * HD_ + 16 + lr];
      pc1 = wmma_f32(wa, vb, pc1);
      vb.x = Vp[(size_t)(kk + 0) * HD_ + 32 + lr]; vb.y = Vp[(size_t)(kk + 1) * HD_ + 32 + lr];
      pc2 = wmma_f32(wa, vb, pc2);
      vb.x = Vp[(size_t)(kk + 0) * HD_ + 48 + lr]; vb.y = Vp[(size_t)(kk + 1) * HD_ + 48 + lr];
      pc3 = wmma_f32(wa, vb, pc3);
    }
#pragma unroll
    for (int i = 0; i < 8; ++i) {
      const int ro = (i + 8 * half) * 64 + lr;
      atomicAdd(&acc[ro +  0], pc0[i]);   // ds_add_f32 cross-wave reduce
      atomicAdd(&acc[ro + 16], pc1[i]);
      atomicAdd(&acc[ro + 32], pc2[i]);
      atomicAdd(&acc[ro + 48], pc3[i]);
    }
    __syncthreads();

    for (int idx = tid; idx < 16 * 64; idx += 256) {
      const int rr = idx >> 6, d = idx & 63;
      ctx[(size_t)((t0 + rr) * B_ + bb) * E_ + h * HD_ + d] = acc[idx];
    }
    __syncthreads();   // Sw/acc reusable next head
  }
}

extern "C" void kernel_launch(void* const* d_in, const int* in_sizes, int n_in,
                              void* d_out, int out_size, void* d_ws, size_t ws_size,
                              hipStream_t stream) {
  const float* x     = (const float*)d_in[0];
  const float* in_w  = (const float*)d_in[1];
  const float* in_b  = (const float*)d_in[2];
  const float* out_w = (const float*)d_in[3];
  const float* out_b = (const float*)d_in[4];

  float* attn = (float*)d_out;                       // [T,B,E]
  float* avgW = attn + (size_t)T_ * B_ * E_;         // [B,T,T]

  const size_t per = (size_t)B_ * H_ * T_ * HD_;     // 4,194,304 floats
  float* Q   = (float*)d_ws;
  float* K   = Q + per;
  float* V   = K + per;
  float* ctx = V + per;                              // [T*B, E]

  // 1) QKV projection + bias + q-scale, scattered to per-head layout
  dim3 g1((3 * E_) / 64, (T_ * B_) / 64);
  gemm_xwT_kernel<<<g1, 128, 0, stream>>>(x, in_w, in_b, nullptr, Q, K, V, 3 * E_, E_, 1);

  // 2) attention (scores, softmax, averaged weights, P*V)
  const size_t smem = (size_t)(16 * SWP + 16 * 64) * sizeof(float);  // ~135 KB of the 320 KB LDS
  attn_kernel<<<B_ * (T_ / 16), 256, smem, stream>>>(Q, K, V, ctx, avgW);

  // 3) output projection
  dim3 g3(E_ / 64, (T_ * B_) / 64);
  gemm_xwT_kernel<<<g3, 128, 0, stream>>>(ctx, out_w, out_b, attn, nullptr, nullptr, nullptr, E_, E_, 0);
}